// SO3net_34737695490102
// MI455X (gfx1250) — compile-verified
//
#include <hip/hip_runtime.h>
#include <hip/hip_bf16.h>
#include <math.h>

// ---------------------------------------------------------------------------
// Types for CDNA5 WMMA (fp32 path: V_WMMA_F32_16X16X4_F32)
// ---------------------------------------------------------------------------
typedef float v2f __attribute__((ext_vector_type(2)));
typedef float v8f __attribute__((ext_vector_type(8)));

#define NSH    9
#define FDIM   64
#define NRBF   20
#define NPAIR  10000
#define NATOM  1000
#define XLEN   (NATOM * NSH * FDIM)   // 576000
#define MROWS  (NATOM * NSH)          // 9000
#define MTILES ((MROWS + 15) / 16)    // 563
#define PROWS  (MTILES * 16)          // 9008 (padded rows for workspace mats)
#define PLEN   (PROWS * FDIM)         // 576512

// ---------------------------------------------------------------------------
// Device helpers for building the real Clebsch-Gordan table (lmax = 2)
// ---------------------------------------------------------------------------
__device__ __forceinline__ int lof(int s) { return (s == 0) ? 0 : ((s < 4) ? 1 : 2); }
__device__ __forceinline__ int mof(int s) {
    switch (s) {
        case 0: return 0;
        case 1: return -1; case 2: return 0; case 3: return 1;
        case 4: return -2; case 5: return -1; case 6: return 0; case 7: return 1; default: return 2;
    }
}
__device__ double dfact(int n) { double r = 1.0; for (int i = 2; i <= n; ++i) r *= (double)i; return r; }

__device__ double cg_complex_d(int a, int b, int c) {
    int l1 = lof(a), m1 = mof(a), l2 = lof(b), m2 = mof(b), l3 = lof(c), m3 = mof(c);
    if (m3 != m1 + m2) return 0.0;
    int dl = l1 - l2; if (dl < 0) dl = -dl;
    if (l3 < dl || l3 > l1 + l2) return 0.0;
    double pre = sqrt((2.0 * l3 + 1.0) * dfact(l3 + l1 - l2) * dfact(l3 - l1 + l2) *
                      dfact(l1 + l2 - l3) / dfact(l1 + l2 + l3 + 1));
    pre *= sqrt(dfact(l3 + m3) * dfact(l3 - m3) * dfact(l1 - m1) * dfact(l1 + m1) *
                dfact(l2 - m2) * dfact(l2 + m2));
    double s = 0.0;
    for (int k = 0; k <= l1 + l2 - l3; ++k) {
        int d1 = l1 + l2 - l3 - k, d2 = l1 - m1 - k, d3 = l2 + m2 - k;
        int d4 = l3 - l2 + m1 + k, d5 = l3 - l1 - m2 + k;
        if (d1 < 0 || d2 < 0 || d3 < 0 || d4 < 0 || d5 < 0) continue;
        double denom = dfact(k) * dfact(d1) * dfact(d2) * dfact(d3) * dfact(d4) * dfact(d5);
        s += ((k & 1) ? -1.0 : 1.0) / denom;
    }
    return pre * s;
}

__device__ void ucoef(int r, int c, double* re, double* im) {
    *re = 0.0; *im = 0.0;
    int lr = lof(r), lc = lof(c);
    if (lr != lc) return;
    int base = lr * lr + lr;
    int mr = r - base, mc = c - base;
    const double is2 = 0.70710678118654752440;
    if (mr == 0) { if (mc == 0) *re = 1.0; return; }
    if (mr > 0) {
        if (mc == -mr)      *re = is2;
        else if (mc == mr)  *re = ((mr & 1) ? -is2 : is2);
    } else {
        int m = -mr;
        if (mc == mr)       *im = is2;
        else if (mc == m)   *im = ((m & 1) ? is2 : -is2);
    }
}
__device__ void ucols(int r, int* c0, int* c1, int* nc) {
    int l = lof(r), m = mof(r), base = l * l + l;
    if (m == 0) { *c0 = r; *c1 = r; *nc = 1; }
    else { int am = (m < 0) ? -m : m; *c0 = base - am; *c1 = base + am; *nc = 2; }
}

__global__ void build_cg_k(float* __restrict__ cgraw) {
    for (int idx = threadIdx.x; idx < 729; idx += blockDim.x) {
        int a = idx / 81, b = (idx / 9) % 9, c = idx % 9;
        int ai[2], bi[2], ci[2], na, nb, nc;
        ucols(a, &ai[0], &ai[1], &na);
        ucols(b, &bi[0], &bi[1], &nb);
        ucols(c, &ci[0], &ci[1], &nc);
        double re = 0.0;
        for (int ii = 0; ii < na; ++ii)
            for (int jj = 0; jj < nb; ++jj)
                for (int kk = 0; kk < nc; ++kk) {
                    int i = ai[ii], j = bi[jj], k = ci[kk];
                    double g = cg_complex_d(i, j, k);
                    if (g == 0.0) continue;
                    double ur, ui, vr, vi, wr, wi;
                    ucoef(a, i, &ur, &ui);
                    ucoef(b, j, &vr, &vi);
                    ucoef(c, k, &wr, &wi);
                    wi = -wi;                              // conj
                    double tr = ur * vr - ui * vi;
                    double ti = ur * vi + ui * vr;
                    double sr = tr * wr - ti * wi;
                    re += g * sr;
                }
        int pa = lof(a) & 1, pb = lof(b) & 1, pc = lof(c) & 1;
        bool mask = ((pa ^ pb) == pc);
        cgraw[c * 81 + a * 9 + b] = mask ? (float)re : 0.0f;   // transposed (c,a,b)
    }
}

__global__ void compact_cg_k(const float* __restrict__ cgraw, int* __restrict__ cnt,
                             int* __restrict__ s1a, int* __restrict__ s2a,
                             int* __restrict__ s3a, int* __restrict__ wia,
                             float* __restrict__ cva) {
    if (threadIdx.x != 0 || blockIdx.x != 0) return;
    int n = 0;
    for (int idx = 0; idx < 729; ++idx) {
        float v = cgraw[idx];
        if (fabsf(v) > 1e-9f) {
            int s1 = idx / 81, s2 = (idx / 9) % 9, s3 = idx % 9;
            s1a[n] = s1; s2a[n] = s2; s3a[n] = s3; wia[n] = lof(s3); cva[n] = v;
            ++n;
        }
    }
    cnt[0] = n;
}

// ---------------------------------------------------------------------------
// x init: x[:,0,:] = emb[Z], rest 0
// ---------------------------------------------------------------------------
__global__ void init_x_k(float* __restrict__ x, const float* __restrict__ emb,
                         const int* __restrict__ Z) {
    int tid = blockIdx.x * blockDim.x + threadIdx.x;
    if (tid >= XLEN) return;
    int a = tid / (NSH * FDIM);
    int r = tid - a * (NSH * FDIM);
    int sh = r >> 6, f = r & 63;
    x[tid] = (sh == 0) ? emb[Z[a] * FDIM + f] : 0.0f;
}

__global__ void zero_k(float* __restrict__ p, int n) {
    int tid = blockIdx.x * blockDim.x + threadIdx.x;
    if (tid < n) p[tid] = 0.0f;
}

// ---------------------------------------------------------------------------
// Pair precompute: Yij (9) and Wij = (rbf @ Wf + bf) * cutoff  (3*64)
// ---------------------------------------------------------------------------
__global__ void pair_wij_k(const float* __restrict__ r_ij, const float* __restrict__ Wf,
                           const float* __restrict__ bf, float* __restrict__ Yij,
                           float* __restrict__ Wij) {
    int lane = threadIdx.x & 31;
    int wv = threadIdx.x >> 5;
    int p = blockIdx.x * 8 + wv;
    if (p >= NPAIR) return;
    float rx = r_ij[3 * p + 0], ry = r_ij[3 * p + 1], rz = r_ij[3 * p + 2];
    float d = sqrtf(rx * rx + ry * ry + rz * rz);
    float inv = 1.0f / d;
    float ux = rx * inv, uy = ry * inv, uz = rz * inv;
    if (lane == 0) {
        const float c0 = 0.28209479177387814f;  // 0.5/sqrt(pi)
        const float c1 = 0.48860251190291992f;  // sqrt(3/(4pi))
        const float c2 = 1.09254843059207907f;  // 0.5*sqrt(15/pi)
        const float c6 = 0.31539156525252005f;  // 0.25*sqrt(5/pi)
        const float c8 = 0.54627421529603953f;  // 0.25*sqrt(15/pi)
        float* Y = Yij + p * NSH;
        Y[0] = c0;           Y[1] = c1 * uy;      Y[2] = c1 * uz;
        Y[3] = c1 * ux;      Y[4] = c2 * ux * uy; Y[5] = c2 * uy * uz;
        Y[6] = c6 * (3.0f * uz * uz - 1.0f);
        Y[7] = c2 * ux * uz; Y[8] = c8 * (ux * ux - uy * uy);
    }
    float radial[NRBF];
    const float width = 5.0f / 19.0f;
    const float coef = -0.5f / (width * width);
#pragma unroll
    for (int r = 0; r < NRBF; ++r) {
        float off = 5.0f * (float)r / 19.0f;
        float t = d - off;
        radial[r] = __expf(coef * t * t);
    }
    float cut = (d < 5.0f) ? 0.5f * (__cosf(d * 0.62831853071795864f) + 1.0f) : 0.0f;
#pragma unroll
    for (int jj = 0; jj < 6; ++jj) {
        int o = lane + 32 * jj;
        float acc = bf[o];
        for (int r = 0; r < NRBF; ++r) acc += radial[r] * Wf[r * 192 + o];
        Wij[p * 192 + o] = acc * cut;
    }
}

// ---------------------------------------------------------------------------
// Messages: per pair CG tensor product, scatter-add into dx[idx_i]
// ---------------------------------------------------------------------------
#define MSG_WSTRIDE 1184   // 16 (Y, padded) + 576 (xj) + 576 (yij) + pad
__global__ void messages_k(const float* __restrict__ x, const int* __restrict__ idx_i,
                           const int* __restrict__ idx_j, const float* __restrict__ Yij,
                           const float* __restrict__ Wij, const int* __restrict__ cgcnt,
                           const int* __restrict__ s1a, const int* __restrict__ s2a,
                           const int* __restrict__ s3a, const int* __restrict__ wia,
                           const float* __restrict__ cva, float* __restrict__ dx) {
    __shared__ float lds[8 * MSG_WSTRIDE];
    int lane = threadIdx.x & 31;
    int wv = threadIdx.x >> 5;
    int p = blockIdx.x * 8 + wv;
    if (p >= NPAIR) return;
    float* LY  = lds + wv * MSG_WSTRIDE;
    float* LXJ = LY + 16;
    float* LYJ = LXJ + 576;
    if (lane < NSH) LY[lane] = Yij[p * NSH + lane];
    int j = idx_j[p];
    const float* xj = x + j * (NSH * FDIM);
#pragma unroll
    for (int s = 0; s < NSH; ++s) {
        LXJ[s * 64 + lane]      = xj[s * 64 + lane];
        LXJ[s * 64 + lane + 32] = xj[s * 64 + lane + 32];
        LYJ[s * 64 + lane]      = 0.0f;
        LYJ[s * 64 + lane + 32] = 0.0f;
    }
    const float* wp = Wij + p * 192;
    float2 Wl0 = { wp[lane],        wp[lane + 32] };
    float2 Wl1 = { wp[64 + lane],   wp[64 + lane + 32] };
    float2 Wl2 = { wp[128 + lane],  wp[128 + lane + 32] };
    int cnt = cgcnt[0];
    for (int e = 0; e < cnt; ++e) {
        int s1 = s1a[e], s2 = s2a[e], s3 = s3a[e], w = wia[e];
        float yc = LY[s3] * cva[e];
        float2 Wv = (w == 0) ? Wl0 : ((w == 1) ? Wl1 : Wl2);
        LYJ[s1 * 64 + lane]      += LXJ[s2 * 64 + lane]      * Wv.x * yc;
        LYJ[s1 * 64 + lane + 32] += LXJ[s2 * 64 + lane + 32] * Wv.y * yc;
    }
    int i = idx_i[p];
    float* dxi = dx + i * (NSH * FDIM);
#pragma unroll
    for (int s = 0; s < NSH; ++s) {
        atomicAdd(dxi + s * 64 + lane,      LYJ[s * 64 + lane]);
        atomicAdd(dxi + s * 64 + lane + 32, LYJ[s * 64 + lane + 32]);
    }
}

// ---------------------------------------------------------------------------
// 64x64 GEMM via V_WMMA_F32_16X16X4_F32.
// A/D are padded to PROWS rows when !ADDC (workspace), so the hot path has no
// divergent bounds checks. B is staged into LDS once per block (shared by the
// 4 waves = 4 N-tiles). ADDC fuses the residual D = A@B + Cin on Mvalid rows;
// only the last block takes the (uniform-branch) guarded path.
// ---------------------------------------------------------------------------
template <bool ADDC>
__global__ void mm64_wmma_k(const float* __restrict__ A, const float* __restrict__ B,
                            const float* __restrict__ Cin, float* __restrict__ D,
                            int Mvalid) {
    __shared__ float Bs[64 * 64];
    int tid = threadIdx.x;                 // 128 threads
    {
        const float4* Bv = (const float4*)B;
        float4* Bsv = (float4*)Bs;
#pragma unroll
        for (int it = 0; it < 8; ++it) Bsv[tid + it * 128] = Bv[tid + it * 128];
    }
    __syncthreads();

    int lane = tid & 31;
    int wv = tid >> 5;                     // 0..3 -> N tile
    int n0 = wv * 16;
    int row0 = blockIdx.x * 16;
    int r = lane & 15;
    int half = lane >> 4;
    bool full = !ADDC || (row0 + 16 <= Mvalid);   // uniform across the block

    v8f c;
    if (ADDC) {
        if (full) {
#pragma unroll
            for (int i = 0; i < 8; ++i)
                c[i] = Cin[(row0 + i + 8 * half) * 64 + n0 + r];
        } else {
#pragma unroll
            for (int i = 0; i < 8; ++i) {
                int row = row0 + i + 8 * half;
                c[i] = (row < Mvalid) ? Cin[row * 64 + n0 + r] : 0.0f;
            }
        }
    } else {
#pragma unroll
        for (int i = 0; i < 8; ++i) c[i] = 0.0f;
    }

    const float* ap = A + (row0 + r) * 64 + 2 * half;   // A rows are padded: no guard
#pragma unroll
    for (int k0 = 0; k0 < 64; k0 += 4) {
        v2f a;
        a[0] = ap[k0 + 0];
        a[1] = ap[k0 + 1];
        v2f b;
        b[0] = Bs[(k0 + 2 * half) * 64 + n0 + r];
        b[1] = Bs[(k0 + 2 * half + 1) * 64 + n0 + r];
        c = __builtin_amdgcn_wmma_f32_16x16x4_f32(false, a, false, b, (short)0, c,
                                                  false, false);
    }

    if (full) {
#pragma unroll
        for (int i = 0; i < 8; ++i)
            D[(row0 + i + 8 * half) * 64 + n0 + r] = c[i];
    } else {
#pragma unroll
        for (int i = 0; i < 8; ++i) {
            int row = row0 + i + 8 * half;
            if (row < Mvalid) D[row * 64 + n0 + r] = c[i];
        }
    }
}

// ---------------------------------------------------------------------------
// Self tensor product: u = dx + CG(dx, ddx). One wave per atom, LDS tiles.
// ---------------------------------------------------------------------------
#define TP_WSTRIDE 1760
__global__ void tp_u_k(const float* __restrict__ dx, const float* __restrict__ ddx,
                       const int* __restrict__ cgcnt, const int* __restrict__ s1a,
                       const int* __restrict__ s2a, const int* __restrict__ s3a,
                       const float* __restrict__ cva, float* __restrict__ u) {
    __shared__ float lds[8 * TP_WSTRIDE];
    int lane = threadIdx.x & 31;
    int wv = threadIdx.x >> 5;
    int a = blockIdx.x * 8 + wv;
    if (a >= NATOM) return;
    float* LDX = lds + wv * TP_WSTRIDE;
    float* LDD = LDX + 576;
    float* LTP = LDD + 576;
    const float* dxa = dx + a * (NSH * FDIM);
    const float* dda = ddx + a * (NSH * FDIM);
#pragma unroll
    for (int s = 0; s < NSH; ++s) {
        LDX[s * 64 + lane]      = dxa[s * 64 + lane];
        LDX[s * 64 + lane + 32] = dxa[s * 64 + lane + 32];
        LDD[s * 64 + lane]      = dda[s * 64 + lane];
        LDD[s * 64 + lane + 32] = dda[s * 64 + lane + 32];
        LTP[s * 64 + lane]      = 0.0f;
        LTP[s * 64 + lane + 32] = 0.0f;
    }
    int cnt = cgcnt[0];
    for (int e = 0; e < cnt; ++e) {
        int s1 = s1a[e], s2 = s2a[e], s3 = s3a[e];
        float cv = cva[e];
        LTP[s1 * 64 + lane]      += LDX[s2 * 64 + lane]      * LDD[s3 * 64 + lane]      * cv;
        LTP[s1 * 64 + lane + 32] += LDX[s2 * 64 + lane + 32] * LDD[s3 * 64 + lane + 32] * cv;
    }
    float* ua = u + a * (NSH * FDIM);
#pragma unroll
    for (int s = 0; s < NSH; ++s) {
        ua[s * 64 + lane]      = LDX[s * 64 + lane]      + LTP[s * 64 + lane];
        ua[s * 64 + lane + 32] = LDX[s * 64 + lane + 32] + LTP[s * 64 + lane + 32];
    }
}

// ---------------------------------------------------------------------------
// Gate: g = sigmoid(dx2[:,0,:] @ Wg + bg); dx2 *= g[LIDX]   (in place)
// ---------------------------------------------------------------------------
__global__ void gate_k(float* __restrict__ dx2, const float* __restrict__ Wg,
                       const float* __restrict__ bg) {
    int lane = threadIdx.x & 31;
    int wv = threadIdx.x >> 5;
    int a = blockIdx.x * 8 + wv;
    if (a >= NATOM) return;
    const float* x0 = dx2 + a * (NSH * FDIM);   // sh = 0 row
    float acc[6];
#pragma unroll
    for (int jj = 0; jj < 6; ++jj) acc[jj] = bg[lane + 32 * jj];
    for (int k = 0; k < 64; ++k) {
        float xv = x0[k];
        const float* wrow = Wg + k * 192;
#pragma unroll
        for (int jj = 0; jj < 6; ++jj) acc[jj] += xv * wrow[lane + 32 * jj];
    }
    float sig[6];
#pragma unroll
    for (int jj = 0; jj < 6; ++jj) sig[jj] = 1.0f / (1.0f + __expf(-acc[jj]));
    const int lsh[NSH] = {0, 1, 1, 1, 2, 2, 2, 2, 2};
    float* da = dx2 + a * (NSH * FDIM);
#pragma unroll
    for (int s = 0; s < NSH; ++s) {
        int l = lsh[s];
        da[s * 64 + lane]      *= sig[2 * l];
        da[s * 64 + lane + 32] *= sig[2 * l + 1];
    }
}

// ---------------------------------------------------------------------------
// Launcher
// ---------------------------------------------------------------------------
extern "C" void kernel_launch(void* const* d_in, const int* in_sizes, int n_in,
                              void* d_out, int out_size, void* d_ws, size_t ws_size,
                              hipStream_t stream) {
    (void)in_sizes; (void)n_in; (void)out_size; (void)ws_size;
    const int*   Z     = (const int*)d_in[0];
    const float* r_ij  = (const float*)d_in[1];
    const int*   idx_i = (const int*)d_in[2];
    const int*   idx_j = (const int*)d_in[3];
    const float* emb   = (const float*)d_in[4];
    const float* Wf    = (const float*)d_in[5];
    const float* bf    = (const float*)d_in[6];
    const float* W1    = (const float*)d_in[7];
    const float* W2    = (const float*)d_in[8];
    const float* W3    = (const float*)d_in[9];
    const float* Wg    = (const float*)d_in[10];
    const float* bg    = (const float*)d_in[11];
    float* x = (float*)d_out;

    float* wsf = (float*)d_ws;
    float* cgraw = wsf + 0;             // 729
    int*   cgcnt = (int*)(wsf + 736);   // 1
    int*   s1a   = (int*)(wsf + 768);
    int*   s2a   = (int*)(wsf + 1536);
    int*   s3a   = (int*)(wsf + 2304);
    int*   wia   = (int*)(wsf + 3072);
    float* cva   = wsf + 3840;
    float* Yij   = wsf + 4608;          // 90000
    float* Wij   = wsf + 94608;         // 1920000
    float* dx    = wsf + 2014608;       // PLEN = 576512 (padded to 9008 rows)
    float* ddx   = wsf + 2591120;       // PLEN
    float* u     = wsf + 3167632;       // PLEN
    float* dx2   = wsf + 3744144;       // PLEN   (end: 4320656 floats ~ 17.3 MB)

    build_cg_k<<<1, 256, 0, stream>>>(cgraw);
    compact_cg_k<<<1, 1, 0, stream>>>(cgraw, cgcnt, s1a, s2a, s3a, wia, cva);
    init_x_k<<<(XLEN + 255) / 256, 256, 0, stream>>>(x, emb, Z);
    // zero u's pad rows once so mm2's padded A reads are defined
    zero_k<<<(PLEN + 255) / 256, 256, 0, stream>>>(u, PLEN);

    for (int t = 0; t < 2; ++t) {
        pair_wij_k<<<(NPAIR + 7) / 8, 256, 0, stream>>>(r_ij, Wf + t * NRBF * 192,
                                                        bf + t * 192, Yij, Wij);
        zero_k<<<(PLEN + 255) / 256, 256, 0, stream>>>(dx, PLEN);
        messages_k<<<(NPAIR + 7) / 8, 256, 0, stream>>>(x, idx_i, idx_j, Yij, Wij,
                                                        cgcnt, s1a, s2a, s3a, wia, cva, dx);
        mm64_wmma_k<false><<<MTILES, 128, 0, stream>>>(dx, W1 + t * 4096, nullptr,
                                                       ddx, MROWS);
        tp_u_k<<<(NATOM + 7) / 8, 256, 0, stream>>>(dx, ddx, cgcnt, s1a, s2a, s3a, cva, u);
        mm64_wmma_k<false><<<MTILES, 128, 0, stream>>>(u, W2 + t * 4096, nullptr,
                                                       dx2, MROWS);
        gate_k<<<(NATOM + 7) / 8, 256, 0, stream>>>(dx2, Wg + t * 64 * 192, bg + t * 192);
        mm64_wmma_k<true><<<MTILES, 128, 0, stream>>>(dx2, W3 + t * 4096, x, x, MROWS);
    }
}